// Parser_9801115369544
// MI455X (gfx1250) — compile-verified
//
#include <hip/hip_runtime.h>
#include <hip/hip_bf16.h>
#include <stdint.h>

// ---------------------------------------------------------------------------
// PYSEQM "parser" for MI455X (gfx1250).
// Store-bandwidth-bound (~405 MB of pair outputs): one workgroup per molecule,
// coords/species staged via CDNA5 async global->LDS DMA (ASYNCcnt), all big
// pair outputs written with non-temporal hints (output > 192MB L2).
// ---------------------------------------------------------------------------

#define NMOL     2048
#define MOLSIZE  100
#define PPM      4950                       // MOLSIZE*(MOLSIZE-1)/2
#define LCF      1.8897261258369282f

static constexpr size_t kNAtoms  = (size_t)NMOL * MOLSIZE;       // 204800
static constexpr size_t kNPairs  = (size_t)NMOL * PPM;           // 10137600

// Output layout (floats, concatenated in reference return order)
static constexpr size_t O_NSUPER = 0;
static constexpr size_t O_NHEAVY = O_NSUPER + NMOL;
static constexpr size_t O_NHYDRO = O_NHEAVY + NMOL;
static constexpr size_t O_NOCC   = O_NHYDRO + NMOL;
static constexpr size_t O_Z      = O_NOCC   + NMOL;
static constexpr size_t O_MASKD  = O_Z      + kNAtoms;
static constexpr size_t O_AMOLID = O_MASKD  + kNAtoms;
static constexpr size_t O_MASK   = O_AMOLID + kNAtoms;
static constexpr size_t O_PMOLID = O_MASK   + kNPairs;
static constexpr size_t O_NI     = O_PMOLID + kNPairs;
static constexpr size_t O_NJ     = O_NI     + kNPairs;
static constexpr size_t O_IDXI   = O_NJ     + kNPairs;
static constexpr size_t O_IDXJ   = O_IDXI   + kNPairs;
static constexpr size_t O_XIJ    = O_IDXJ   + kNPairs;
static constexpr size_t O_RIJ    = O_XIJ    + 3 * kNPairs;

// CDNA5 async DMA: global -> LDS, 16B per lane, tracked by ASYNCcnt.
// Low 32 bits of a generic pointer to __shared__ data are the LDS byte
// address (ISA 10.2, LDS aperture).
__device__ __forceinline__ void async_load_b128(void* lds_dst, const void* gsrc) {
  uint32_t lds_off = (uint32_t)(uintptr_t)lds_dst;
  asm volatile("global_load_async_to_lds_b128 %0, %1, off"
               :: "v"(lds_off), "v"(gsrc)
               : "memory");
}
__device__ __forceinline__ void wait_async_zero() {
  asm volatile("s_wait_asynccnt 0" ::: "memory");
}

__device__ __forceinline__ void nts(float* p, float v) {
  __builtin_nontemporal_store(v, p);
}

__device__ __forceinline__ int tri_off(int i) {           // #pairs with row < i
  return (i * (2 * MOLSIZE - 1 - i)) >> 1;                // i*(199-i)/2
}

__global__ __launch_bounds__(256)
void parser_kernel(const float* __restrict__ coordinates,
                   const float* __restrict__ tore,
                   const int*   __restrict__ species,
                   const int*   __restrict__ tot_charge,
                   float*       __restrict__ out)
{
  __shared__ __align__(16) float s_coord[MOLSIZE * 3];    // 1200 B
  __shared__ __align__(16) int   s_spec[MOLSIZE];         // 400 B
  __shared__ float s_csum;
  __shared__ int   s_heavy, s_super, s_hydro;

  const int mol = blockIdx.x;
  const int tid = threadIdx.x;

  const float* gc = coordinates + (size_t)mol * (MOLSIZE * 3);
  const int*   gs = species     + (size_t)mol * MOLSIZE;

  if (tid == 0) { s_csum = 0.0f; s_heavy = 0; s_super = 0; s_hydro = 0; }

  // Stage this molecule's coords (300 dwords = 75 x b128) and species
  // (100 dwords = 25 x b128) through the async LDS path.
  if (tid < 100) {
    if (tid < 75) async_load_b128(&s_coord[tid * 4],        gc + tid * 4);
    else          async_load_b128(&s_spec[(tid - 75) * 4],  gs + (tid - 75) * 4);
  }
  wait_async_zero();          // each wave drains its own ASYNCcnt
  __syncthreads();            // then LDS visible block-wide

  // ---- per-atom outputs + per-molecule reductions -------------------------
  if (tid < MOLSIZE) {
    int z = s_spec[tid];
    int heavy = (z > 1);
    int hydro = (z == 1);
    int super_ = ((z > 12 && z < 18) || (z > 20 && z < 30) ||
                  (z > 32 && z < 36) || (z > 38 && z < 48) ||
                  (z > 50 && z < 54) || (z > 70 && z < 80) || (z == 57));
    if (heavy)  atomicAdd(&s_heavy, 1);
    if (hydro)  atomicAdd(&s_hydro, 1);
    if (super_) atomicAdd(&s_super, 1);
    atomicAdd(&s_csum, tore[z]);

    size_t a = (size_t)mol * MOLSIZE + tid;
    out[O_Z + a]      = (float)z;
    out[O_MASKD + a]  = (float)(tid * (MOLSIZE + 1) + mol * MOLSIZE * MOLSIZE);
    out[O_AMOLID + a] = (float)mol;
  }
  __syncthreads();

  if (tid == 0) {
    out[O_NSUPER + mol] = (float)s_super;
    out[O_NHEAVY + mol] = (float)s_heavy;
    out[O_NHYDRO + mol] = (float)s_hydro;
    int ncharge = (int)s_csum - tot_charge[mol];
    out[O_NOCC + mol]   = (float)(ncharge / 2);
  }

  // ---- all i<j pairs of this molecule (4950), NT streaming stores ---------
  for (int k = tid; k < PPM; k += 256) {
    // closed-form row from triangular index, then exact fixup
    float disc = 39601.0f - 8.0f * (float)k;               // (2M-1)^2 - 8k
    int i = (int)(0.5f * (199.0f - sqrtf(disc)));
    if (i < 0) i = 0;
    if (i > MOLSIZE - 2) i = MOLSIZE - 2;
    while (tri_off(i + 1) <= k) ++i;
    while (tri_off(i) > k) --i;
    int j = k - tri_off(i) + i + 1;

    float dx = s_coord[3 * j + 0] - s_coord[3 * i + 0];
    float dy = s_coord[3 * j + 1] - s_coord[3 * i + 1];
    float dz = s_coord[3 * j + 2] - s_coord[3 * i + 2];
    float d2 = dx * dx + dy * dy + dz * dz;
    float rinv = rsqrtf(d2);
    float dist = d2 * rinv;

    size_t p  = (size_t)mol * PPM + k;
    int   ai  = mol * MOLSIZE + i;
    int   aj  = mol * MOLSIZE + j;

    nts(out + O_MASK   + p, (float)(ai * MOLSIZE + j));    // (b*100+i)*100 + j
    nts(out + O_PMOLID + p, (float)mol);
    nts(out + O_NI     + p, (float)s_spec[i]);
    nts(out + O_NJ     + p, (float)s_spec[j]);
    nts(out + O_IDXI   + p, (float)ai);
    nts(out + O_IDXJ   + p, (float)aj);

    float* x = out + O_XIJ + 3 * p;                        // per-lane b96
    nts(x + 0, dx * rinv);
    nts(x + 1, dy * rinv);
    nts(x + 2, dz * rinv);

    nts(out + O_RIJ + p, dist * LCF);
  }
}

extern "C" void kernel_launch(void* const* d_in, const int* in_sizes, int n_in,
                              void* d_out, int out_size, void* d_ws, size_t ws_size,
                              hipStream_t stream) {
  const float* coordinates = (const float*)d_in[0];   // (2048,100,3) f32
  const float* tore        = (const float*)d_in[1];   // (86,) f32
  const int*   species     = (const int*)d_in[2];     // (2048,100) i32
  const int*   tot_charge  = (const int*)d_in[3];     // (2048,) i32
  float*       out         = (float*)d_out;

  parser_kernel<<<NMOL, 256, 0, stream>>>(coordinates, tore, species, tot_charge, out);
}